// Bilinear_21311627723279
// MI455X (gfx1250) — compile-verified
//
#include <hip/hip_runtime.h>
#include <cstdint>

// Bilinear gather-sample on MI455X (gfx1250).
// Memory-bound op: WMMA is inapplicable (no matmul structure). CDNA5 features
// used instead: async global->LDS coordinate staging (ASYNCcnt, double
// buffered, s_wait_asynccnt) and non-temporal output stores to keep the
// gather working set (entire 128 MB input) resident in the 192 MB L2.

namespace {
constexpr int kH = 224;
constexpr int kW = 224;
constexpr int kB = 128;
constexpr int kHW = kH * kW;            // 50176
constexpr int kNPix = kB * kHW;         // 6,422,528
constexpr int kTPB = 256;               // 8 wave32 waves
constexpr int kNChunk = kNPix / kTPB;   // 25,088 (exact -> no tail masking)
constexpr int kNBlk = 2048;             // grid-stride: ~12 chunks per block
}  // namespace

// Stage one lane's (X,Y) pair (bytes 12..19 of the 20-byte pixel) into LDS
// via the CDNA5 async memory engine. GVS mode: saddr(64) + vaddr(i32) + ioff.
__device__ __forceinline__ void async_coords(uint32_t ldsX, uint32_t ldsY,
                                             const float* xbase,
                                             uint32_t byteOff) {
  asm volatile(
      "global_load_async_to_lds_b32 %0, %2, %3 offset:12\n\t"
      "global_load_async_to_lds_b32 %1, %2, %3 offset:16"
      :
      : "v"(ldsX), "v"(ldsY), "v"(byteOff), "s"(xbase)
      : "memory");
}

__global__ __launch_bounds__(kTPB) void bilinear_kernel(
    const float* __restrict__ x, float* __restrict__ out) {
  __shared__ float sX[2][kTPB];
  __shared__ float sY[2][kTPB];
  const int tid = threadIdx.x;

  // LDS byte offsets for this lane's slots (generic-ptr low 32 bits == LDS off)
  const uint32_t xoff[2] = {(uint32_t)(uintptr_t)&sX[0][tid],
                            (uint32_t)(uintptr_t)&sX[1][tid]};
  const uint32_t yoff[2] = {(uint32_t)(uintptr_t)&sY[0][tid],
                            (uint32_t)(uintptr_t)&sY[1][tid]};

  int chunk = (int)blockIdx.x;
  // Prologue: stage first chunk's coords into buffer 0.
  async_coords(xoff[0], yoff[0], x, ((uint32_t)chunk * kTPB + tid) * 20u);

  int buf = 0;
  for (; chunk < kNChunk; chunk += kNBlk) {
    // Prefetch next chunk's coords into the other buffer (clamped: harmless
    // refetch of the current chunk on the last iteration; always in-bounds).
    const int next = chunk + kNBlk;
    const int pre = (next < kNChunk) ? next : chunk;
    async_coords(xoff[buf ^ 1], yoff[buf ^ 1], x,
                 ((uint32_t)pre * kTPB + tid) * 20u);

    // Exactly the 2 just-issued async ops may remain outstanding; async loads
    // complete in order, so <=2 guarantees the current buffer is in LDS.
    asm volatile("s_wait_asynccnt 0x2" ::: "memory");

    const float Xc = sX[buf][tid];
    const float Yc = sY[buf][tid];

    const float fx = floorf(Xc);
    const float fy = floorf(Yc);
    const float wx = Xc - fx;
    const float wy = Yc - fy;
    const float omx = 1.0f - wx;
    const float omy = 1.0f - wy;
    const float w_tl = omx * omy;
    const float w_bl = omx * wy;
    const float w_tr = wx * omy;
    const float w_br = wx * wy;

    const int p = chunk * kTPB + tid;
    const int b = p / kHW;  // batch index (const-div -> mul/shift)

    // Clamp-to-edge tap indices (equivalent to reference's pad+clip for the
    // actual coordinate range [0,223)).
    int ix = (int)fx, iy = (int)fy;
    const int c0 = min(max(ix, 0), kW - 1);
    const int c1 = min(max(ix + 1, 0), kW - 1);
    const int r0 = min(max(iy, 0), kH - 1);
    const int r1 = min(max(iy + 1, 0), kH - 1);

    const float* base = x + (size_t)b * (size_t)kHW * 5u;
    const float* t00 = base + (size_t)(r0 * kW + c0) * 5u;  // tl
    const float* t01 = base + (size_t)(r0 * kW + c1) * 5u;  // tr
    const float* t10 = base + (size_t)(r1 * kW + c0) * 5u;  // bl
    const float* t11 = base + (size_t)(r1 * kW + c1) * 5u;  // br

    // Reference summation order: tl + bl + tr + br.
    const float o0 = w_tl * t00[0] + w_bl * t10[0] + w_tr * t01[0] + w_br * t11[0];
    const float o1 = w_tl * t00[1] + w_bl * t10[1] + w_tr * t01[1] + w_br * t11[1];
    const float o2 = w_tl * t00[2] + w_bl * t10[2] + w_tr * t01[2] + w_br * t11[2];

    // Stream the 77 MB output non-temporally: don't evict the gather working
    // set (whole input fits in the 192 MB L2).
    float* op = out + (size_t)p * 3u;
    __builtin_nontemporal_store(o0, op + 0);
    __builtin_nontemporal_store(o1, op + 1);
    __builtin_nontemporal_store(o2, op + 2);

    buf ^= 1;
  }
  // Drain the dangling prefetch before wave exit (S_ENDPGM waits anyway).
  asm volatile("s_wait_asynccnt 0x0" ::: "memory");
}

extern "C" void kernel_launch(void* const* d_in, const int* in_sizes, int n_in,
                              void* d_out, int out_size, void* d_ws,
                              size_t ws_size, hipStream_t stream) {
  (void)in_sizes; (void)n_in; (void)out_size; (void)d_ws; (void)ws_size;
  const float* xp = (const float*)d_in[0];
  float* op = (float*)d_out;
  hipLaunchKernelGGL(bilinear_kernel, dim3(kNBlk), dim3(kTPB), 0, stream, xp, op);
}